// SingleSamplePreintegrationModule_4681514353468
// MI455X (gfx1250) — compile-verified
//
#include <hip/hip_runtime.h>
#include <math.h>

// ---------------------------------------------------------------------------
// IMU preintegration, elementwise over N samples.
//   in : [N,6] f32 (w, a)   out : [N,10] f32 (delta_p, quat, delta_v)
// Memory-bound: 64 B/sample, 256 MB total -> ~11 us at 23.3 TB/s.
// WMMA is inapplicable (per-sample 3x3 matrices, no shared GEMM dimension).
// CDNA5 angle: stage tiles through LDS with async global<->LDS b128 DMA ops
// (ASYNCcnt path) so all HBM traffic is dense 16B/lane transfers despite the
// 24B/40B per-sample strides.
// ---------------------------------------------------------------------------

#define DT 0.005f
#define EPSF 1e-12f

constexpr int THREADS = 256;   // 8 x wave32
constexpr int TILE    = 512;   // samples per block

typedef float v4f __attribute__((ext_vector_type(4)));
typedef int   v4i __attribute__((ext_vector_type(4)));

#define AS1 __attribute__((address_space(1)))
#define AS3 __attribute__((address_space(3)))

#if defined(__HIP_DEVICE_COMPILE__) && \
    __has_builtin(__builtin_amdgcn_global_load_async_to_lds_b128) && \
    __has_builtin(__builtin_amdgcn_global_store_async_from_lds_b128)
#define USE_ASYNC 1
#else
#define USE_ASYNC 0
#endif

__device__ static inline void wait_async0() {
#if __has_builtin(__builtin_amdgcn_s_wait_asynccnt)
  __builtin_amdgcn_s_wait_asynccnt(0);
#else
  asm volatile("s_wait_asynccnt 0" ::: "memory");
#endif
}

// Core per-sample math (matches reference formulas).
__device__ static inline void preint_sample(const float wx, const float wy, const float wz,
                                            const float ax, const float ay, const float az,
                                            float* __restrict__ o /* 10 floats */) {
  // axis-angle -> rotation matrix (Rodrigues)
  const float n2   = wx * wx + wy * wy + wz * wz + EPSF;
  const float nrm  = __builtin_sqrtf(n2);
  const float inv  = 1.0f / nrm;
  const float th   = nrm * DT;
  const float x = wx * inv, y = wy * inv, z = wz * inv;

  float st, ct;
  __sincosf(th, &st, &ct);
  const float c = 1.0f - ct;

  const float xx = x * x, yy = y * y, zz = z * z;
  const float xy = x * y, xz = x * z, yz = y * z;

  const float R00 = 1.0f - c * (yy + zz);
  const float R01 = -st * z + c * xy;
  const float R02 =  st * y + c * xz;
  const float R10 =  st * z + c * xy;
  const float R11 = 1.0f - c * (xx + zz);
  const float R12 = -st * x + c * yz;
  const float R20 = -st * y + c * xz;
  const float R21 =  st * x + c * yz;
  const float R22 = 1.0f - c * (xx + yy);

  // delta_v = R @ (a*dt)
  const float avx = ax * DT, avy = ay * DT, avz = az * DT;
  const float dvx = R00 * avx + R01 * avy + R02 * avz;
  const float dvy = R10 * avx + R11 * avy + R12 * avz;
  const float dvz = R20 * avx + R21 * avy + R22 * avz;

  // delta_p = dv*dt + R@(a*dt^2/2) == dv * 1.5*dt
  const float pf = 1.5f * DT;
  const float dpx = dvx * pf, dpy = dvy * pf, dpz = dvz * pf;

  // rotmat -> axis-angle -> quaternion
  const float trace = R00 + R11 + R22;
  float cos_t = (trace - 1.0f) * 0.5f;
  cos_t = fminf(fmaxf(cos_t, -1.0f + 1e-7f), 1.0f - 1e-7f);
  const float th2  = acosf(cos_t);
  const float s2   = __builtin_sqrtf(fmaxf(1.0f - cos_t * cos_t, 0.0f)); // sin(acos(x))
  const float den  = 1.0f / (2.0f * s2 + EPSF);
  const float vx = (R21 - R12) * den;
  const float vy = (R02 - R20) * den;
  const float vz = (R10 - R01) * den;

  float sh, ch;
  __sincosf(0.5f * th2, &sh, &ch);

  o[0] = dpx;     o[1] = dpy;     o[2] = dpz;
  o[3] = ch;      o[4] = vx * sh; o[5] = vy * sh; o[6] = vz * sh;
  o[7] = dvx;     o[8] = dvy;     o[9] = dvz;
}

__global__ __launch_bounds__(THREADS)
void preint_kernel(const float* __restrict__ in, float* __restrict__ out, long long N) {
  __shared__ float sIn[TILE * 6];    // 12288 B
  __shared__ float sOut[TILE * 10];  // 20480 B

  const int t = threadIdx.x;
  const long long s0 = (long long)blockIdx.x * TILE;

  // ---- tail block: plain per-sample path (N is actually a multiple of TILE) ----
  if (s0 + TILE > N) {
    for (long long i = s0 + t; i < N; i += THREADS) {
      const float* g = in + i * 6;
      float o[10];
      preint_sample(g[0], g[1], g[2], g[3], g[4], g[5], o);
      float* go = out + i * 10;
#pragma unroll
      for (int j = 0; j < 10; ++j) go[j] = o[j];
    }
    return;
  }

  const float* gin  = in  + s0 * 6;   // 16B-aligned chunk (512*24 B)
  float*       gout = out + s0 * 10;  // 16B-aligned chunk (512*40 B)

  // ---- stage input tile into LDS: 768 x b128, 3 per lane ----
#if USE_ASYNC
#pragma unroll
  for (int k = 0; k < 3; ++k) {
    const int idx = t + k * THREADS;  // float4 index
    __builtin_amdgcn_global_load_async_to_lds_b128(
        (AS1 v4i*)(gin + idx * 4), (AS3 v4i*)(sIn + idx * 4), 0, 0);
  }
  wait_async0();
#else
#pragma unroll
  for (int k = 0; k < 3; ++k) {
    const int idx = t + k * THREADS;
    *(v4f*)(sIn + idx * 4) = *(const v4f*)(gin + idx * 4);
  }
#endif
  __syncthreads();

  // ---- compute: 2 samples per thread, LDS -> LDS ----
#pragma unroll
  for (int k = 0; k < 2; ++k) {
    const int i = t + k * THREADS;
    const float* s = sIn + i * 6;
    preint_sample(s[0], s[1], s[2], s[3], s[4], s[5], sOut + i * 10);
  }
  __syncthreads();

  // ---- drain output tile: 1280 x b128, 5 per lane ----
#if USE_ASYNC
#pragma unroll
  for (int k = 0; k < 5; ++k) {
    const int idx = t + k * THREADS;
    __builtin_amdgcn_global_store_async_from_lds_b128(
        (AS1 v4i*)(gout + idx * 4), (AS3 v4i*)(sOut + idx * 4), 0, 0);
  }
  // S_ENDPGM performs an implicit wait-idle; LDS stays allocated until then.
#else
#pragma unroll
  for (int k = 0; k < 5; ++k) {
    const int idx = t + k * THREADS;
    *(v4f*)(gout + idx * 4) = *(const v4f*)(sOut + idx * 4);
  }
#endif
}

extern "C" void kernel_launch(void* const* d_in, const int* in_sizes, int n_in,
                              void* d_out, int out_size, void* d_ws, size_t ws_size,
                              hipStream_t stream) {
  const float* in  = (const float*)d_in[0];
  float*       out = (float*)d_out;
  const long long total = in_sizes[0];     // N * 6
  const long long N     = total / 6;
  const int blocks = (int)((N + TILE - 1) / TILE);
  preint_kernel<<<blocks, THREADS, 0, stream>>>(in, out, N);
}